// ISTFT_87351044866856
// MI455X (gfx1250) — compile-verified
//
#include <hip/hip_runtime.h>

typedef float v2f __attribute__((ext_vector_type(2)));
typedef float v8f __attribute__((ext_vector_type(8)));

#define NFFT     1024
#define WIN      1024
#define HOP      256
#define BATCH    8
#define TFRAMES  2000
#define FBINS    513                  // NFFT/2 + 1
#define KDIM     1026                 // 2*FBINS
#define KCHUNK   32                   // K rows staged per LDS chunk (16 K-pair segments)
#define KPAD     1056                 // K padded to multiple of KCHUNK (33 chunks)
#define NCHUNKS  (KPAD / KCHUNK)      // 33
#define BLK_M    128                  // rows per block (8 waves x 16)
#define BLK_N    64                   // cols per block
#define NSEG     (KCHUNK / 2)         // 16 K-pair segments per chunk
#define SEGSTRIDE (2 * BLK_N + 32)    // 160 floats: 128 data + 32 pad (bank split)
#define M_TOTAL  (BATCH * TFRAMES)    // 16000
#define L_FULL   ((TFRAMES - 1) * HOP + WIN)  // 512768
#define OUT_PER_B (L_FULL - WIN)      // 511744

// ---------------------------------------------------------------------------
// Async global->LDS copy (GLOBAL_LOAD_ASYNC_TO_LDS_B128, ASYNCcnt) via inline
// asm (portable; the clang builtin's param types use the unspellable HIP
// __device__ LangAS). VDST = per-lane LDS byte offset, VADDR = 64-bit global
// address, GV mode.
// ---------------------------------------------------------------------------
__device__ __forceinline__ void async_copy_b128(const void* gaddr, void* lds_ptr) {
    unsigned lds_off = (unsigned)(unsigned long long)
        (__attribute__((address_space(3))) void*)lds_ptr;   // 32-bit LDS offset
    unsigned long long g = (unsigned long long)gaddr;
    asm volatile("global_load_async_to_lds_b128 %0, %1, off"
                 :
                 : "v"(lds_off), "v"(g)
                 : "memory");
}

__device__ __forceinline__ void wait_async_zero() {
#if defined(__has_builtin)
#if __has_builtin(__builtin_amdgcn_s_wait_asynccnt)
    __builtin_amdgcn_s_wait_asynccnt(0);
#else
    asm volatile("s_wait_asynccnt 0" ::: "memory");
#endif
#else
    asm volatile("s_wait_asynccnt 0" ::: "memory");
#endif
}

// ---------------------------------------------------------------------------
// Kernel 1: windowed inverse-DFT basis in K-PAIR-INTERLEAVED layout:
//   element (k, n) -> basis[(k/2)*2048 + 2*n + (k%2)]
// so the WMMA B operand (rows 2i, 2i+1 at one column) is two consecutive
// floats = one ds_load_b64 after LDS staging.
//   k <  513 :  ck*cos(2*pi*k*n/1024)/1024 * window[n]
//   k < 1026 : -ck*sin(2*pi*(k-513)*n/1024)/1024 * window[n]
//   k >=1026 :  0  (K padding)
// ---------------------------------------------------------------------------
__global__ __launch_bounds__(256) void build_basis(const float* __restrict__ window,
                                                   float* __restrict__ basis) {
    int idx = blockIdx.x * blockDim.x + threadIdx.x;
    if (idx >= KPAD * NFFT) return;
    int k = idx >> 10;
    int n = idx & (NFFT - 1);
    float val = 0.0f;
    if (k < KDIM) {
        int kk = (k < FBINS) ? k : (k - FBINS);
        float ck = (kk == 0 || kk == FBINS - 1) ? 1.0f : 2.0f;
        int m = (kk * n) & (NFFT - 1);                      // exact periodic reduction
        float phase = (float)m * 6.1359231515425647e-03f;   // 2*pi/1024
        float tr = (k < FBINS) ? cosf(phase) : -sinf(phase);
        val = ck * tr * (1.0f / (float)NFFT) * window[n];
    }
    basis[((size_t)(k >> 1) << 11) + ((size_t)n << 1) + (k & 1)] = val;
}

// ---------------------------------------------------------------------------
// Kernel 2: sig[16000][1024] = Z^T[M][K] * basis[K][N] via V_WMMA_F32_16X16X4_F32.
// Block = 256 threads (8 waves) -> 128(M) x 64(N) tile. Basis chunks
// (32K x 64N, interleaved, 10KB w/ pad) double-buffered in LDS via async b128
// copies (ASYNCcnt), shared by all 8 waves.
//
// B fragments: one ds_load_b64 each, ALL from a single per-wave base VGPR
// (lb + half*SEGSTRIDE + 2*lm) with immediate offsets (k4*2*SEGSTRIDE + 32j)*4
// <= 9984B (fits 16-bit DS offset). Segment pad (+32 floats) puts the odd
// half-wave on banks 32..63 -> conflict-free b64 reads.
//
// Fragment layouts (ISA 7.12.2, 32-bit, wave32):
//   A 16x4: lanes 0-15 M=lane (K0/K1), lanes 16-31 M=lane-16 (K2/K3)
//   B 4x16: same half-wave K split, N = lane%16
//   C/D:    VGPR r holds M=r (lanes 0-15) / M=r+8 (lanes 16-31), N = lane%16
// ---------------------------------------------------------------------------
__global__ __launch_bounds__(256) void istft_gemm(const float* __restrict__ z,
                                                  const float* __restrict__ basis,
                                                  float* __restrict__ sig) {
    __shared__ alignas(16) float bbuf[2][NSEG * SEGSTRIDE];   // 2 x 10 KB

    const int tid  = threadIdx.x;
    const int lane = tid & 31;
    const int wv   = tid >> 5;               // 0..7
    const int ngrp = blockIdx.x;             // 0..15  (N block of 64)
    const int mblk = blockIdx.y;             // 0..124

    const int m0   = mblk * BLK_M + wv * 16; // wave's 16-row base (multiple of 16)
    const int b    = m0 / TFRAMES;           // 2000 % 16 == 0: wave never crosses batch
    const int t0   = m0 % TFRAMES;
    const int half = lane >> 4;              // 0 / 1
    const int lm   = lane & 15;
    const int koff0 = half * 2;              // A K-offset: 0 or 2 (pairs are k, k+1)

    const float* __restrict__ zA = z + (size_t)b * KDIM * TFRAMES + t0 + lm;
    // Interleaved basis: segment s (K rows 2s,2s+1) at s*2048, our 64 columns
    // start at 2*ngrp*64 within the segment.
    const float* __restrict__ gB = basis + (size_t)ngrp * (2 * BLK_N);

    // Stage one chunk (16 segments x 128 floats) into bbuf[buf]; 512 float4
    // items, 2 per thread; global segment stride 2048, LDS stride SEGSTRIDE.
    auto stage = [&](int chunk, int buf) {
        const float* src = gB + ((size_t)(chunk * NSEG) << 11);
#pragma unroll
        for (int j = 0; j < 2; ++j) {
            int item = tid + j * 256;        // 0..511
            int seg  = item >> 5;            // 0..15
            int f4   = (item & 31) << 2;     // float offset within segment: 0..124
            async_copy_b128(src + ((size_t)seg << 11) + f4,
                            &bbuf[buf][seg * SEGSTRIDE + f4]);
        }
    };

    v8f acc0 = {}, acc1 = {}, acc2 = {}, acc3 = {};

    // One K4-step: A rows (ka0, ka0+1) for this half-wave from z; B pairs from
    // LDS with constant offsets off one shared base.
    auto k4_step = [&](const float* lbase, int k4, int ka0, int ka1) {
        v2f a;
        a.x = zA[(size_t)ka0 * TFRAMES];
        a.y = zA[(size_t)ka1 * TFRAMES];

        const float* s = lbase + k4 * (2 * SEGSTRIDE);
        v2f b0 = *(const v2f*)(s + 0);
        v2f b1 = *(const v2f*)(s + 32);
        v2f b2 = *(const v2f*)(s + 64);
        v2f b3 = *(const v2f*)(s + 96);

        acc0 = __builtin_amdgcn_wmma_f32_16x16x4_f32(false, a, false, b0, (short)0, acc0, false, false);
        acc1 = __builtin_amdgcn_wmma_f32_16x16x4_f32(false, a, false, b1, (short)0, acc1, false, false);
        acc2 = __builtin_amdgcn_wmma_f32_16x16x4_f32(false, a, false, b2, (short)0, acc2, false, false);
        acc3 = __builtin_amdgcn_wmma_f32_16x16x4_f32(false, a, false, b3, (short)0, acc3, false, false);
    };

    stage(0, 0);
    wait_async_zero();
    __syncthreads();

    // Chunks 0..NCHUNKS-2: no K clamping needed (max k = 1023 < 1026).
    for (int c = 0; c < NCHUNKS - 1; ++c) {
        const int p = c & 1;
        stage(c + 1, p ^ 1);                        // prefetch next chunk (other buffer)
        const int kbase = c * KCHUNK;
        const float* lbase = &bbuf[p][0] + half * SEGSTRIDE + 2 * lm;  // per-wave const
#pragma unroll
        for (int k4 = 0; k4 < KCHUNK / 4; ++k4) {
            const int ka = kbase + k4 * 4 + koff0;
            k4_step(lbase, k4, ka, ka + 1);
        }
        wait_async_zero();  // chunk c+1's async copies have landed
        __syncthreads();    // all waves done reading buffer p / writes visible
    }

    {   // Final chunk (kbase = 1024): clamp A indices; pad basis rows are zero.
        const int p = (NCHUNKS - 1) & 1;
        const int kbase = (NCHUNKS - 1) * KCHUNK;
        const float* lbase = &bbuf[p][0] + half * SEGSTRIDE + 2 * lm;
#pragma unroll
        for (int k4 = 0; k4 < KCHUNK / 4; ++k4) {
            int ka0 = kbase + k4 * 4 + koff0;
            int ka1 = ka0 + 1;
            if (ka0 > KDIM - 1) ka0 = KDIM - 1;
            if (ka1 > KDIM - 1) ka1 = KDIM - 1;
            k4_step(lbase, k4, ka0, ka1);
        }
    }

    // Store D tiles: row (half*8 + r), columns ngrp*64 + {0,16,32,48} + lm
    float* __restrict__ out = sig + (size_t)m0 * NFFT + ngrp * BLK_N + lm;
    const int mrow = half * 8;
#pragma unroll
    for (int r = 0; r < 8; ++r) {
        float* orow = out + (size_t)(mrow + r) * NFFT;
        orow[0]  = acc0[r];
        orow[16] = acc1[r];
        orow[32] = acc2[r];
        orow[48] = acc3[r];
    }
}

// ---------------------------------------------------------------------------
// Kernel 3: overlap-add as a gather (<=4 frames per output sample) + window
// normalization computed on the fly, trimmed output region q in [512, L-512).
// ---------------------------------------------------------------------------
__global__ __launch_bounds__(256) void ola_gather(const float* __restrict__ sig,
                                                  const float* __restrict__ window,
                                                  float* __restrict__ out) {
    int idx = blockIdx.x * blockDim.x + threadIdx.x;
    if (idx >= BATCH * OUT_PER_B) return;
    int b = idx / OUT_PER_B;
    int p = idx - b * OUT_PER_B;
    int q = p + WIN / 2;

    int tmax = q >> 8;                       // floor(q/256)
    if (tmax > TFRAMES - 1) tmax = TFRAMES - 1;
    int tmin = q - (WIN - HOP);              // ceil((q-1023)/256), clamped at 0
    tmin = tmin > 0 ? (tmin >> 8) : 0;

    const float* __restrict__ sb = sig + (size_t)b * TFRAMES * NFFT;
    float acc = 0.0f, wn = 0.0f;
    for (int t = tmin; t <= tmax; ++t) {
        int n = q - (t << 8);                // in [0, 1024)
        float w = window[n];
        acc += sb[(size_t)t * NFFT + n];
        wn  += w * w;
    }
    out[idx] = acc / wn;
}

// ---------------------------------------------------------------------------
extern "C" void kernel_launch(void* const* d_in, const int* in_sizes, int n_in,
                              void* d_out, int out_size, void* d_ws, size_t ws_size,
                              hipStream_t stream) {
    const float* z      = (const float*)d_in[0];   // [8, 1026, 2000] f32
    const float* window = (const float*)d_in[1];   // [1024] f32
    float* out = (float*)d_out;                    // [8, 511744] f32

    float* basis = (float*)d_ws;                         // KPAD * NFFT f32 (~4.3 MB, interleaved)
    float* sig   = basis + (size_t)KPAD * NFFT;          // M_TOTAL * NFFT f32 (~65.5 MB)

    {   // 1) basis (K-pair interleaved)
        int total  = KPAD * NFFT;
        int blocks = (total + 255) / 256;                // 4224
        build_basis<<<blocks, 256, 0, stream>>>(window, basis);
    }
    {   // 2) WMMA GEMM with LDS-staged (async, double-buffered) basis
        dim3 grid(NFFT / BLK_N, M_TOTAL / BLK_M);        // (16, 125)
        istft_gemm<<<grid, 256, 0, stream>>>(z, basis, sig);
    }
    {   // 3) overlap-add gather + normalize
        int total  = BATCH * OUT_PER_B;
        int blocks = (total + 255) / 256;
        ola_gather<<<blocks, 256, 0, stream>>>(sig, window, out);
    }
}